// GCNBackbone_21277267984978
// MI455X (gfx1250) — compile-verified
//
#include <hip/hip_runtime.h>
#include <hip/hip_bf16.h>
#include <math.h>

// ---------------------------------------------------------------------------
// GCN backbone: 3 x (dense f32 GEMM via V_WMMA_F32_16X16X4_F32 + normalized
// scatter-add over edges with native global_atomic_add_f32), exact-erf GELU.
//
// D_HIDDEN = 128 fixed.
// ---------------------------------------------------------------------------

typedef float v2f __attribute__((ext_vector_type(2)));
typedef float v8f __attribute__((ext_vector_type(8)));

#define DH 128

// ---------------------------------------------------------------------------
// deg[i] = 1.0 (self loop)
__global__ void gcn_deg_init(float* __restrict__ deg, int n) {
    int i = blockIdx.x * blockDim.x + threadIdx.x;
    if (i < n) deg[i] = 1.0f;
}

// deg[dst[e]] += 1
__global__ void gcn_deg_count(const int* __restrict__ dst, float* __restrict__ deg, int ne) {
    int e = blockIdx.x * blockDim.x + threadIdx.x;
    if (e < ne) atomicAdd(&deg[dst[e]], 1.0f);
}

// dinv = rsqrt(deg)   (deg >= 1 always due to self loops)
__global__ void gcn_dinv(float* __restrict__ deg, int n) {
    int i = blockIdx.x * blockDim.x + threadIdx.x;
    if (i < n) deg[i] = rsqrtf(deg[i]);
}

// nrm[e] = dinv[src[e]] * dinv[dst[e]]   (layer-invariant, computed once)
__global__ void gcn_norm(const int* __restrict__ src, const int* __restrict__ dst,
                         const float* __restrict__ dinv, float* __restrict__ nrm, int ne) {
    int e = blockIdx.x * blockDim.x + threadIdx.x;
    if (e < ne) nrm[e] = dinv[src[e]] * dinv[dst[e]];
}

// Wt[l][n][k] = W[l][k][n]   (so GEMM B-operand loads become 8B vector loads)
__global__ void gcn_transpose_w(const float* __restrict__ W, float* __restrict__ Wt, int count) {
    int i = blockIdx.x * blockDim.x + threadIdx.x;
    if (i >= count) return;
    int l = i >> 14;              // layer
    int rem = i & 16383;
    int k = rem >> 7;
    int n = rem & (DH - 1);
    Wt[(l << 14) + (n << 7) + k] = W[i];
}

// ---------------------------------------------------------------------------
// H[m0:m0+16, :] = X[m0:m0+16, :] @ W     via V_WMMA_F32_16X16X4_F32
// grid.x = ceil(nrows/16), block = 256 (8 waves), wave w owns N-tile w (16 cols).
// Wt is column-major W (i.e. Wt[n][k]), so both A and B load as float2.
//
// A (16x4 f32): lanes 0-15 -> M=0..15 {K=k,k+1}; lanes 16-31 -> {K=k+2,k+3}
// B (4x16 f32): lanes 0-15 -> N=0..15 {K=k,k+1}; lanes 16-31 -> {K=k+2,k+3}
// C/D (16x16 f32): VGPR v, lanes 0-15 -> M=v,N=lane; lanes 16-31 -> M=v+8,N=lane-16
__global__ void __launch_bounds__(256)
gcn_gemm(const float* __restrict__ X, const float* __restrict__ Wt,
         float* __restrict__ H, int nrows) {
    const int lane = threadIdx.x & 31;
    const int wv   = threadIdx.x >> 5;     // 0..7 : which 16-col tile
    const int m0   = blockIdx.x << 4;
    const int n0   = wv << 4;
    const int r    = lane & 15;
    const int half = lane >> 4;            // 0 or 1

    int arow = m0 + r;
    if (arow >= nrows) arow = nrows - 1;   // clamp (harmless, stores guarded)

    const float* ap = X  + (size_t)arow     * DH + 2 * half;   // {K, K+1}
    const float* bp = Wt + (size_t)(n0 + r) * DH + 2 * half;   // {K, K+1}

    v8f c = {};
#pragma unroll
    for (int ks = 0; ks < DH / 4; ++ks) {
        v2f a = *(const v2f*)(ap + 4 * ks);    // 8B-aligned
        v2f b = *(const v2f*)(bp + 4 * ks);
        c = __builtin_amdgcn_wmma_f32_16x16x4_f32(
                /*neg_a=*/false, a, /*neg_b=*/false, b,
                /*c_mod=*/(short)0, c, /*reuse_a=*/false, /*reuse_b=*/false);
    }

    float* hp = H + (size_t)m0 * DH + n0 + r;
#pragma unroll
    for (int v = 0; v < 8; ++v) {
        int row = v + 8 * half;
        if (m0 + row < nrows) hp[(size_t)row * DH] = c[v];
    }
}

// ---------------------------------------------------------------------------
// Y[i,c] = bias[c] + dinv[i]^2 * H[i,c]   (bias + folded self-loop message)
// float4-vectorized: one thread per 4 elements.
__global__ void gcn_init_y(float* __restrict__ Y, const float* __restrict__ H,
                           const float* __restrict__ dinv,
                           const float* __restrict__ bias, int total4) {
    int i = blockIdx.x * blockDim.x + threadIdx.x;
    if (i >= total4) return;
    int node = i >> 5;                 // (4*i) / 128
    int col4 = i & 31;                 // float4 column index
    float di = dinv[node];
    di *= di;
    float4 h = ((const float4*)H)[i];
    float4 b = ((const float4*)bias)[col4];
    ((float4*)Y)[i] = make_float4(b.x + di * h.x, b.y + di * h.y,
                                  b.z + di * h.z, b.w + di * h.w);
}

// ---------------------------------------------------------------------------
// One wave per edge. Edge metadata (src, dst, nrm) is wave-uniform ->
// readfirstlane forces scalar s_loads, keeping the VMEM pipe for the
// float4 gather + 4 native global_atomic_add_f32 per lane (512B/edge total).
__global__ void __launch_bounds__(256)
gcn_scatter(const float* __restrict__ H, float* __restrict__ Y,
            const int* __restrict__ src, const int* __restrict__ dst,
            const float* __restrict__ nrm, int nedges) {
    int e = (blockIdx.x << 3) + (threadIdx.x >> 5);
    if (e >= nedges) return;
    e = __builtin_amdgcn_readfirstlane(e);
    int lane = threadIdx.x & 31;

    int   s = src[e];           // scalar loads (uniform address)
    int   d = dst[e];
    float w = nrm[e];

    float4 hv = ((const float4*)(H + (size_t)s * DH))[lane];
    float* yp = Y + (size_t)d * DH + (lane << 2);
    float  m0 = hv.x * w, m1 = hv.y * w, m2 = hv.z * w, m3 = hv.w * w;

    asm volatile("global_atomic_add_f32 %0, %1, off\n\t"
                 "global_atomic_add_f32 %0, %2, off offset:4\n\t"
                 "global_atomic_add_f32 %0, %3, off offset:8\n\t"
                 "global_atomic_add_f32 %0, %4, off offset:12"
                 :: "v"(yp), "v"(m0), "v"(m1), "v"(m2), "v"(m3)
                 : "memory");
}

// ---------------------------------------------------------------------------
// Exact GELU: 0.5*x*(1+erf(x/sqrt(2))), in place, float4-vectorized.
__global__ void gcn_gelu(float* __restrict__ Y, int total4) {
    int i = blockIdx.x * blockDim.x + threadIdx.x;
    if (i >= total4) return;
    float4 v = ((float4*)Y)[i];
    const float k = 0.70710678118654752440f;
    v.x = 0.5f * v.x * (1.0f + erff(v.x * k));
    v.y = 0.5f * v.y * (1.0f + erff(v.y * k));
    v.z = 0.5f * v.z * (1.0f + erff(v.z * k));
    v.w = 0.5f * v.w * (1.0f + erff(v.w * k));
    ((float4*)Y)[i] = v;
}

// ---------------------------------------------------------------------------
extern "C" void kernel_launch(void* const* d_in, const int* in_sizes, int n_in,
                              void* d_out, int out_size, void* d_ws, size_t ws_size,
                              hipStream_t stream) {
    const float* x    = (const float*)d_in[0];     // [N, 128] f32
    const int*   eidx = (const int*)  d_in[1];     // [2, E]   i32
    const float* Ws   = (const float*)d_in[2];     // [3, 128, 128] f32
    const float* bs   = (const float*)d_in[3];     // [3, 128] f32
    float*       out  = (float*)d_out;             // [N, 128] f32

    const int N = in_sizes[0] / DH;    // 50000
    const int E = in_sizes[1] / 2;     // 800000
    const int* src = eidx;             // row 0: message sources
    const int* dst = eidx + E;         // row 1: aggregation targets

    // workspace layout (floats):
    // [dinv : N pad64] [nrm : E pad64] [Wt : 3*128*128] [h : N*128] [buf : N*128]
    float* ws   = (float*)d_ws;
    float* dinv = ws;
    float* nrm  = dinv + (((size_t)N + 63) & ~(size_t)63);
    float* Wt   = nrm  + (((size_t)E + 63) & ~(size_t)63);
    float* h    = Wt   + (size_t)3 * DH * DH;
    float* buf  = h    + (size_t)N * DH;

    const int threads = 256;
    const int total4  = N * DH / 4;
    const int wcount  = 3 * DH * DH;

    // --- degree / normalization (edge structure is layer-invariant) ---
    gcn_deg_init   <<<(N + threads - 1) / threads, threads, 0, stream>>>(dinv, N);
    gcn_deg_count  <<<(E + threads - 1) / threads, threads, 0, stream>>>(dst, dinv, E);
    gcn_dinv       <<<(N + threads - 1) / threads, threads, 0, stream>>>(dinv, N);
    gcn_norm       <<<(E + threads - 1) / threads, threads, 0, stream>>>(src, dst, dinv, nrm, E);
    gcn_transpose_w<<<(wcount + threads - 1) / threads, threads, 0, stream>>>(Ws, Wt, wcount);

    // --- 3 GCN layers ---
    const float* xin = x;
    for (int l = 0; l < 3; ++l) {
        float* y = (l == 2) ? out : buf;
        gcn_gemm   <<<(N + 15) / 16, 256, 0, stream>>>(xin, Wt + (size_t)l * DH * DH, h, N);
        gcn_init_y <<<(total4 + threads - 1) / threads, threads, 0, stream>>>(y, h, dinv, bs + l * DH, total4);
        gcn_scatter<<<(E + 7) / 8, 256, 0, stream>>>(h, y, src, dst, nrm, E);
        xin = y;   // l=0: buf; l=1: gemm consumes buf before it is re-initialized
    }

    // --- exact GELU on final output ---
    gcn_gelu<<<(total4 + threads - 1) / threads, threads, 0, stream>>>(out, total4);
}